// MC_63995012710625
// MI455X (gfx1250) — compile-verified
//
#include <hip/hip_runtime.h>
#include <hip/hip_bf16.h>

// ---------------------------------------------------------------------------
// Star-graph GAT, algebraically collapsed:
//   kernel 1: streaming pass over fp (~21 MB => ~0.9us HBM roofline) computing
//             per (batch, branch): xw = softmax-weighted sum of x rows (7-dim)
//             and x0 = node-0 features. Deterministic LDS tree reductions.
//             Pads each 16-float ws row with zeros at [7] and [15] so kernel 2
//             can load K=8-padded A operands unconditionally.
//   kernel 2: batched dense tail on V_WMMA_F32_16X16X4_F32 (full fp32):
//             H0=xw@W1, H1=x0@W1 -> c = elu(H1)+127*elu(H0)
//             G += c@W2 (both branches) -> out = clamp(G@fc2W + b, 0, 10)
//             All inner-loop operands are unconditional (zero-padded LDS),
//             avoiding exec-mask divergence around loads.
// ---------------------------------------------------------------------------

#define BATCH   2048
#define NNODE   128
#define IN_DIM  7
#define FEAT    10
#define HID     64
#define TAU     22

typedef float v2f __attribute__((ext_vector_type(2)));
typedef float v8f __attribute__((ext_vector_type(8)));

// deterministic block reduction over 128 threads (op 0 = sum, 1 = max)
__device__ __forceinline__ float block_reduce(float v, float* red, int tid, int op) {
    red[tid] = v;
    __syncthreads();
#pragma unroll
    for (int s = 64; s > 0; s >>= 1) {
        if (tid < s) {
            float a0 = red[tid], b0 = red[tid + s];
            red[tid] = op ? fmaxf(a0, b0) : (a0 + b0);
        }
        __syncthreads();
    }
    float r = red[0];
    __syncthreads();
    return r;
}

// ---------------------------------------------------------------------------
// Kernel 1: one block per (batch, branch); 128 threads = 128 nodes.
// ws layout per (b,br): 16 floats: [0..6]=xw, [7]=0, [8..14]=x0, [15]=0.
// ---------------------------------------------------------------------------
__global__ __launch_bounds__(128) void MC_gat_attn(
    const float* __restrict__ fp,
    const float* __restrict__ uW, const float* __restrict__ ua,
    const float* __restrict__ dW, const float* __restrict__ da,
    float* __restrict__ ws)
{
    const int b  = blockIdx.x;
    const int br = blockIdx.y;
    const float* W = br ? dW : uW;   // (7,64) row-major
    const float* a = br ? da : ua;   // (128,1): [0..63]=a_lo, [64..127]=a_hi
    const int tid = threadIdx.x;

    __shared__ float xs[NNODE * FEAT];   // raw feature rows (coalesced load)
    __shared__ float wl[IN_DIM];         // W @ a_lo   (7)
    __shared__ float wh[IN_DIM];         // W @ a_hi   (7)
    __shared__ float red[128];

    const float* base = fp + ((size_t)b * (2 * NNODE) + br * NNODE) * FEAT;
#pragma unroll
    for (int i = 0; i < FEAT; ++i)
        xs[i * 128 + tid] = base[i * 128 + tid];

    if (tid < 2 * IN_DIM) {
        const int k = tid % IN_DIM, sel = tid / IN_DIM;
        const float* av = a + sel * HID;
        float acc = 0.f;
        for (int d2 = 0; d2 < HID; ++d2) acc += W[k * HID + d2] * av[d2];
        if (sel == 0) wl[k] = acc; else wh[k] = acc;
    }
    __syncthreads();

    float x[IN_DIM];
#pragma unroll
    for (int k = 0; k < IN_DIM; ++k) x[k] = xs[tid * FEAT + k];

    float s1 = 0.f, s2 = 0.f;                       // s1 = x0.(W a_lo), same all lanes
#pragma unroll
    for (int k = 0; k < IN_DIM; ++k) { s1 += xs[k] * wl[k]; s2 += x[k] * wh[k]; }

    float e = s1 + s2;
    e = (e >= 0.f) ? e : 0.2f * e;                  // leaky_relu(., 0.2)

    // softmax over nodes j = 1..127 (node 0 masked out of its own row)
    const float em = (tid == 0) ? -3.0e38f : e;
    const float mx = block_reduce(em, red, tid, 1);
    const float p  = (tid == 0) ? 0.f : expf(e - mx);
    const float sum = block_reduce(p, red, tid, 0);
    const float inv = 1.0f / sum;

    float* o = ws + ((size_t)b * 2 + br) * 16;
#pragma unroll
    for (int k = 0; k < IN_DIM; ++k) {
        const float t = block_reduce(p * x[k], red, tid, 0);
        if (tid == 0) o[k] = t * inv;               // xw[k]
    }
    if (tid < IN_DIM) o[8 + tid] = xs[tid];         // x0[k]
    if (tid == 7) { o[7] = 0.f; o[15] = 0.f; }      // K-padding zeros
}

// ---------------------------------------------------------------------------
// Kernel 2: dense tail on WMMA. One wave = 16 batches; 8 waves/block;
// grid = 2048/(16*8) = 16 blocks of 256 threads.
// ---------------------------------------------------------------------------
__global__ __launch_bounds__(256) void MC_gat_tail(
    const float* __restrict__ ws,
    const float* __restrict__ uW1, const float* __restrict__ uW2,
    const float* __restrict__ dW1, const float* __restrict__ dW2,
    const float* __restrict__ fcW, const float* __restrict__ fcb,
    float* __restrict__ out)
{
    __shared__ float cbuf_all[8][16 * 68];  // per-wave 16x64 tile, stride 68 (34816 B)
    __shared__ float w1s[2][8 * HID];       // W1 padded to 8x64, row 7 = 0   (4096 B)
    __shared__ float fcs[HID * 32];         // fc2W padded to 64x32, col>=22=0 (8192 B)

    const int tid  = threadIdx.x;
    const int wid  = tid >> 5;
    const int lane = tid & 31;
    const int half = lane >> 4;             // K-half / M-half selector
    const int mrow = lane & 15;
    float* cbuf = cbuf_all[wid];
    const int m0 = (blockIdx.x * 8 + wid) * 16;   // first batch of this wave's tile

    // ---- cooperative zero-padded weight staging (one-time) ----
#pragma unroll
    for (int idx = 0; idx < 4; ++idx) {           // 2*8*64 = 1024 floats
        const int i = idx * 256 + tid;
        const int wbr = i >> 9, r = (i >> 6) & 7, col = i & 63;
        const float* W1 = wbr ? dW1 : uW1;
        w1s[wbr][r * HID + col] = (r < IN_DIM) ? W1[r * HID + col] : 0.f;
    }
#pragma unroll
    for (int idx = 0; idx < 8; ++idx) {           // 64*32 = 2048 floats
        const int i = idx * 256 + tid;
        const int r = i >> 5, col = i & 31;
        fcs[i] = (col < TAU) ? fcW[r * TAU + col] : 0.f;
    }
    __syncthreads();

    const v8f zero = {};
    v8f G[4];
#pragma unroll
    for (int n = 0; n < 4; ++n) G[n] = zero;

    const float* wsrow = ws + (size_t)(m0 + mrow) * 32;   // 2 branches * 16 floats

    for (int br = 0; br < 2; ++br) {
        const float* W2 = br ? dW2 : uW2;           // (64,64), global (L2-resident)
        const float* w1 = w1s[br];
        const float* wr = wsrow + br * 16;

        // ---- stage 1: H0 = XW @ W1, H1 = X0 @ W1  (M=16, N=64, K=8 padded) ----
        v8f H0[4], H1[4];
#pragma unroll
        for (int n = 0; n < 4; ++n) { H0[n] = zero; H1[n] = zero; }
#pragma unroll
        for (int c = 0; c < 2; ++c) {
            const int k0 = 4 * c + 2 * half;        // A layout: lanes0-15 K0/K1, 16-31 K2/K3
            const int k1 = k0 + 1;
            v2f a0, a1;
            a0.x = wr[k0];      a0.y = wr[k1];      // wr[7] == 0 by construction
            a1.x = wr[8 + k0];  a1.y = wr[8 + k1];  // wr[15] == 0 by construction
#pragma unroll
            for (int n = 0; n < 4; ++n) {
                v2f bb;
                bb.x = w1[k0 * HID + n * 16 + mrow];   // row 7 == 0 by construction
                bb.y = w1[k1 * HID + n * 16 + mrow];
                H0[n] = __builtin_amdgcn_wmma_f32_16x16x4_f32(false, a0, false, bb, (short)0, H0[n], false, false);
                H1[n] = __builtin_amdgcn_wmma_f32_16x16x4_f32(false, a1, false, bb, (short)0, H1[n], false, false);
            }
        }

        // ---- elementwise: c = elu(H1) + 127*elu(H0) -> wave-private LDS tile ----
#pragma unroll
        for (int n = 0; n < 4; ++n) {
#pragma unroll
            for (int r = 0; r < 8; ++r) {           // D layout: M = r + 8*half, N = mrow+16n
                const float h0 = H0[n][r], h1 = H1[n][r];
                const float v0 = (h0 > 0.f) ? h0 : (expf(h0) - 1.f);
                const float v1 = (h1 > 0.f) ? h1 : (expf(h1) - 1.f);
                cbuf[(r + 8 * half) * 68 + n * 16 + mrow] = v1 + 127.f * v0;
            }
        }

        // ---- stage 2: G += C @ W2  (K = 64) ----
#pragma unroll
        for (int kc = 0; kc < 16; ++kc) {
            const int k0 = 4 * kc + 2 * half;
            v2f aa;
            aa.x = cbuf[mrow * 68 + k0];
            aa.y = cbuf[mrow * 68 + k0 + 1];
#pragma unroll
            for (int n = 0; n < 4; ++n) {
                v2f bb;
                bb.x = W2[k0 * HID + n * 16 + mrow];
                bb.y = W2[(k0 + 1) * HID + n * 16 + mrow];
                G[n] = __builtin_amdgcn_wmma_f32_16x16x4_f32(false, aa, false, bb, (short)0, G[n], false, false);
            }
        }
    }

    // ---- stage 3: O = G @ fc2W  (K=64, N padded to 32) ----
#pragma unroll
    for (int n = 0; n < 4; ++n)
#pragma unroll
        for (int r = 0; r < 8; ++r)
            cbuf[(r + 8 * half) * 68 + n * 16 + mrow] = G[n][r];

    v8f O[2]; O[0] = zero; O[1] = zero;
#pragma unroll
    for (int kc = 0; kc < 16; ++kc) {
        const int k0 = 4 * kc + 2 * half;
        v2f aa;
        aa.x = cbuf[mrow * 68 + k0];
        aa.y = cbuf[mrow * 68 + k0 + 1];
#pragma unroll
        for (int n = 0; n < 2; ++n) {
            const int col = n * 16 + mrow;
            v2f bb;
            bb.x = fcs[k0 * 32 + col];              // cols >= 22 are zero
            bb.y = fcs[(k0 + 1) * 32 + col];
            O[n] = __builtin_amdgcn_wmma_f32_16x16x4_f32(false, aa, false, bb, (short)0, O[n], false, false);
        }
    }

    // ---- bias + relu + clip(0,10) -> out (B, 22) ----
#pragma unroll
    for (int n = 0; n < 2; ++n) {
        const int col = n * 16 + mrow;
        if (col < TAU) {
            const float bias = fcb[col];
#pragma unroll
            for (int r = 0; r < 8; ++r) {
                const int M = r + 8 * half;
                float t = O[n][r] + bias;
                t = fminf(fmaxf(t, 0.f), 10.f);
                out[(size_t)(m0 + M) * TAU + col] = t;
            }
        }
    }
}

extern "C" void kernel_launch(void* const* d_in, const int* in_sizes, int n_in,
                              void* d_out, int out_size, void* d_ws, size_t ws_size,
                              hipStream_t stream) {
    const float* fp    = (const float*)d_in[0];
    const float* uattW = (const float*)d_in[1];
    const float* uatta = (const float*)d_in[2];
    const float* uoutW = (const float*)d_in[3];
    // d_in[4] (u_out_a) mathematically drops out (uniform layer-2 softmax)
    const float* dattW = (const float*)d_in[5];
    const float* datta = (const float*)d_in[6];
    const float* doutW = (const float*)d_in[7];
    // d_in[8] (d_out_a) likewise unused
    const float* fcW   = (const float*)d_in[9];
    const float* fcb   = (const float*)d_in[10];
    float* out = (float*)d_out;
    float* ws  = (float*)d_ws;     // needs BATCH*2*16 floats = 256 KB

    dim3 g1(BATCH, 2);
    MC_gat_attn<<<g1, 128, 0, stream>>>(fp, uattW, uatta, dattW, datta, ws);

    dim3 g2(BATCH / (16 * 8));     // 16 blocks, 8 waves each, 16 batches/wave
    MC_gat_tail<<<g2, 256, 0, stream>>>(ws, uattW, uoutW, dattW, doutW, fcW, fcb, out);
}